// AltPredictor_70342974374025
// MI455X (gfx1250) — compile-verified
//
#include <hip/hip_runtime.h>
#include <math.h>

typedef __attribute__((ext_vector_type(2))) float v2f;
typedef __attribute__((ext_vector_type(8))) float v8f;

#define NB 64
#define NS 300
#define NR 100
#define RP 112          // reads padded to 7 tiles of 16
#define NT 7
#define NOUT (NB * NR)

// Branchless ELU: max(v,0) + (exp(min(v,0)) - 1).  Exact for v>0 since __expf(0)==1.
__device__ __forceinline__ float eluf(float v) {
    return fmaxf(v, 0.f) + (__expf(fminf(v, 0.f)) - 1.f);
}

// monotone float <-> uint mapping for atomicMax over signed floats
__device__ __forceinline__ unsigned fmapu(float f) {
    unsigned u = __float_as_uint(f);
    return (u & 0x80000000u) ? ~u : (u | 0x80000000u);
}
__device__ __forceinline__ float funmap(unsigned u) {
    return __uint_as_float((u & 0x80000000u) ? (u & 0x7fffffffu) : ~u);
}

__device__ __forceinline__ v8f wmma4(v8f c, v2f a, v2f b) {
    // D = A(16x4,f32) * B(4x16,f32) + C(16x16,f32)
    return __builtin_amdgcn_wmma_f32_16x16x4_f32(false, a, false, b, (short)0, c, false, false);
}

// Build B operand fragment: B[k][n] = W[n][colofs + k], zero-padded.
__device__ __forceinline__ v2f makeB(const float* W, int n, int k0, int kmax,
                                     int colofs, int ld, int nmax) {
    v2f b;
    b.x = (n < nmax && k0     < kmax) ? W[n * ld + colofs + k0    ] : 0.f;
    b.y = (n < nmax && k0 + 1 < kmax) ? W[n * ld + colofs + k0 + 1] : 0.f;
    return b;
}

__global__ __launch_bounds__(32)
void alt_predictor_main(const float* __restrict__ x,
                        const float* __restrict__ l1_w, const float* __restrict__ l1_b,
                        const float* __restrict__ l2_w, const float* __restrict__ l2_b,
                        const float* __restrict__ l3_w, const float* __restrict__ l3_b,
                        const float* __restrict__ fc1_w, const float* __restrict__ fc1_b,
                        const float* __restrict__ fc2_w, const float* __restrict__ fc2_b,
                        const float* __restrict__ fc3_w, const float* __restrict__ fc3_b,
                        unsigned* __restrict__ wsmax)
{
    __shared__ float Alds[RP][16];   // col 0 = r, cols 1..10 = x feats, 11..15 = 0
    __shared__ float Ylds[RP][8];    // intermediate activations (5 used, K-padded to 8)
    __shared__ float Hlds[16][8];    // per-tile fc activations
    __shared__ float Ymean[8];

    const int lane = threadIdx.x;
    const int n    = lane & 15;      // N index (B/D) and M index (A)
    const int hi   = lane >> 4;      // lane half selects K/M sub-block
    const int koff = hi * 2;

    const int bs   = blockIdx.x;                 // (b,s) pair
    const int bidx = bs / NS;
    const float* xb = x + (size_t)bs * (NR * 10);

    // ---- zero LDS ----
    for (int i = lane; i < RP * 16; i += 32) (&Alds[0][0])[i] = 0.f;
    for (int i = lane; i < RP * 8;  i += 32) (&Ylds[0][0])[i] = 0.f;
    __syncthreads();

    // ---- load x (coalesced) ----
    for (int i = lane; i < NR * 10; i += 32) Alds[i / 10][1 + (i % 10)] = xb[i];
    __syncthreads();

    // ---- r[m] = dot(x[m,0:4], x[0,0:4]) ----
    const float x00 = Alds[0][1], x01 = Alds[0][2], x02 = Alds[0][3], x03 = Alds[0][4];
    for (int m = lane; m < RP; m += 32)
        Alds[m][0] = Alds[m][1]*x00 + Alds[m][2]*x01 + Alds[m][3]*x02 + Alds[m][4]*x03;

    // ---- b_mean over reads (wave reduction) ----
    float s0 = 0.f, s1 = 0.f, s2 = 0.f, s3 = 0.f;
    for (int m = lane; m < NR; m += 32) {
        s0 += Alds[m][1]; s1 += Alds[m][2]; s2 += Alds[m][3]; s3 += Alds[m][4];
    }
    for (int off = 16; off; off >>= 1) {
        s0 += __shfl_xor(s0, off); s1 += __shfl_xor(s1, off);
        s2 += __shfl_xor(s2, off); s3 += __shfl_xor(s3, off);
    }
    const float bm0 = s0 * (1.f / NR), bm1 = s1 * (1.f / NR),
                bm2 = s2 * (1.f / NR), bm3 = s3 * (1.f / NR);
    __syncthreads();

    // ---- fold row-constant features into biases (zero for n>=5 keeps lanes inert) ----
    float bias1 = 0.f, bias2 = 0.f, bias3 = 0.f, biasf2 = 0.f;
    if (n < 5) {
        bias1 = l1_b[n] + bm0*l1_w[n*15+0] + bm1*l1_w[n*15+1]
                        + bm2*l1_w[n*15+2] + bm3*l1_w[n*15+3];
        bias2 = l2_b[n]; bias3 = l3_b[n]; biasf2 = fc2_b[n];
    }
    const float biasf3 = (n == 0) ? fc3_b[0] : 0.f;

    // ---- precompute B fragments (shared by all tiles) ----
    v2f B1[3], Bf1[3], B2[2], B3[2], Bf2[2], Bf3[2];
#pragma unroll
    for (int st = 0; st < 3; ++st) {
        const int k0 = 4*st + koff;
        B1[st]  = makeB(l1_w,  n, k0, 11, 4, 15, 5);  // eff input [r, x0..x9]
        Bf1[st] = makeB(fc1_w, n, k0, 11, 9, 20, 5);
    }
#pragma unroll
    for (int st = 0; st < 2; ++st) {
        const int k0 = 4*st + koff;
        B2[st]  = makeB(l2_w,  n, k0, 5, 0, 5, 5);
        B3[st]  = makeB(l3_w,  n, k0, 5, 0, 5, 5);
        Bf2[st] = makeB(fc2_w, n, k0, 5, 0, 5, 5);
        Bf3[st] = makeB(fc3_w, n, k0, 5, 0, 5, 1);
    }

    // ================= Phase A: l1 -> l2 -> l3, accumulate ysum =================
    // For lanes with n>=5: B fragments and biases are 0 => acc=0, elu(0)=0, so
    // values stay identically zero through all layers without any masking.
    float ysum = 0.f;
    for (int t = 0; t < NT; ++t) {
        const int arow = t*16 + n;
        v8f acc = {};
#pragma unroll
        for (int st = 0; st < 3; ++st) {
            v2f a; a.x = Alds[arow][4*st + koff]; a.y = Alds[arow][4*st + koff + 1];
            acc = wmma4(acc, a, B1[st]);
        }
        float y0v[8];
#pragma unroll
        for (int v = 0; v < 8; ++v) y0v[v] = eluf(acc[v] + bias1);
        if (n < 8) {
#pragma unroll
            for (int v = 0; v < 8; ++v) Ylds[t*16 + v + 8*hi][n] = y0v[v];
        }
        __syncthreads();

        v8f acc2 = {};
#pragma unroll
        for (int st = 0; st < 2; ++st) {
            v2f a; a.x = Ylds[arow][4*st + koff]; a.y = Ylds[arow][4*st + koff + 1];
            acc2 = wmma4(acc2, a, B2[st]);
        }
        float y1v[8];
#pragma unroll
        for (int v = 0; v < 8; ++v) y1v[v] = eluf(acc2[v] + bias2 + y0v[v]);
        __syncthreads();                 // all y0 reads complete before overwrite
        if (n < 8) {
#pragma unroll
            for (int v = 0; v < 8; ++v) Ylds[t*16 + v + 8*hi][n] = y1v[v];
        }
        __syncthreads();

        v8f acc3 = {};
#pragma unroll
        for (int st = 0; st < 2; ++st) {
            v2f a; a.x = Ylds[arow][4*st + koff]; a.y = Ylds[arow][4*st + koff + 1];
            acc3 = wmma4(acc3, a, B3[st]);
        }
#pragma unroll
        for (int v = 0; v < 8; ++v) {
            const float y2 = eluf(acc3[v] + bias3 + y1v[v]);
            const int gm = t*16 + v + 8*hi;
            ysum += (gm < NR) ? y2 : 0.f;
        }
    }

    // ---- ymean & fc1 effective bias (folds b_mean and ymean terms) ----
    ysum += __shfl_xor(ysum, 16);
    if (lane < 5) Ymean[lane] = ysum * (1.f / NR);
    __syncthreads();
    float biasf1 = 0.f;
    if (n < 5) {
        biasf1 = fc1_b[n]
               + bm0*fc1_w[n*20+5] + bm1*fc1_w[n*20+6] + bm2*fc1_w[n*20+7] + bm3*fc1_w[n*20+8]
               + Ymean[0]*fc1_w[n*20+0] + Ymean[1]*fc1_w[n*20+1] + Ymean[2]*fc1_w[n*20+2]
               + Ymean[3]*fc1_w[n*20+3] + Ymean[4]*fc1_w[n*20+4];
    }

    // ================= Phase B: fc1 -> fc2 -> fc3, atomic max over s =================
    unsigned* wsb = wsmax + bidx * NR;
    for (int t = 0; t < NT; ++t) {
        const int arow = t*16 + n;
        v8f acc = {};
#pragma unroll
        for (int st = 0; st < 3; ++st) {
            v2f a; a.x = Alds[arow][4*st + koff]; a.y = Alds[arow][4*st + koff + 1];
            acc = wmma4(acc, a, Bf1[st]);
        }
        float h0v[8];
#pragma unroll
        for (int v = 0; v < 8; ++v) h0v[v] = eluf(acc[v] + biasf1);
        __syncthreads();                 // prev tile's Hlds reads complete
        if (n < 8) {
#pragma unroll
            for (int v = 0; v < 8; ++v) Hlds[v + 8*hi][n] = h0v[v];
        }
        __syncthreads();

        v8f acc2 = {};
#pragma unroll
        for (int st = 0; st < 2; ++st) {
            v2f a; a.x = Hlds[n][4*st + koff]; a.y = Hlds[n][4*st + koff + 1];
            acc2 = wmma4(acc2, a, Bf2[st]);
        }
        float h1v[8];
#pragma unroll
        for (int v = 0; v < 8; ++v) h1v[v] = eluf(acc2[v] + biasf2 + h0v[v]);
        __syncthreads();
        if (n < 8) {
#pragma unroll
            for (int v = 0; v < 8; ++v) Hlds[v + 8*hi][n] = h1v[v];
        }
        __syncthreads();

        v8f acc3 = {};
#pragma unroll
        for (int st = 0; st < 2; ++st) {
            v2f a; a.x = Hlds[n][4*st + koff]; a.y = Hlds[n][4*st + koff + 1];
            acc3 = wmma4(acc3, a, Bf3[st]);
        }
        if (n == 0) {
#pragma unroll
            for (int v = 0; v < 8; ++v) {
                const int gm = t*16 + v + 8*hi;
                if (gm < NR) {
                    const float h2 = 5.f * eluf(acc3[v] + biasf3);
                    atomicMax(&wsb[gm], fmapu(h2));
                }
            }
        }
    }
}

__global__ void alt_predictor_init(unsigned* __restrict__ w) {
    const int i = blockIdx.x * blockDim.x + threadIdx.x;
    if (i < NOUT) w[i] = 0u;    // below every mapped real float
}

__global__ void alt_predictor_finish(const unsigned* __restrict__ w, float* __restrict__ out) {
    const int i = blockIdx.x * blockDim.x + threadIdx.x;
    if (i < NOUT) {
        const float h = funmap(w[i]);
        out[i] = 1.f / (1.f + __expf(-h));
    }
}

extern "C" void kernel_launch(void* const* d_in, const int* in_sizes, int n_in,
                              void* d_out, int out_size, void* d_ws, size_t ws_size,
                              hipStream_t stream) {
    const float* x     = (const float*)d_in[0];
    const float* l1_w  = (const float*)d_in[1];
    const float* l1_b  = (const float*)d_in[2];
    const float* l2_w  = (const float*)d_in[3];
    const float* l2_b  = (const float*)d_in[4];
    const float* l3_w  = (const float*)d_in[5];
    const float* l3_b  = (const float*)d_in[6];
    const float* fc1_w = (const float*)d_in[7];
    const float* fc1_b = (const float*)d_in[8];
    const float* fc2_w = (const float*)d_in[9];
    const float* fc2_b = (const float*)d_in[10];
    const float* fc3_w = (const float*)d_in[11];
    const float* fc3_b = (const float*)d_in[12];

    unsigned* wsu = (unsigned*)d_ws;

    alt_predictor_init<<<(NOUT + 255) / 256, 256, 0, stream>>>(wsu);
    alt_predictor_main<<<NB * NS, 32, 0, stream>>>(
        x, l1_w, l1_b, l2_w, l2_b, l3_w, l3_b,
        fc1_w, fc1_b, fc2_w, fc2_b, fc3_w, fc3_b, wsu);
    alt_predictor_finish<<<(NOUT + 255) / 256, 256, 0, stream>>>(wsu, (float*)d_out);
}